// VectorMapNet_46454366274162
// MI455X (gfx1250) — compile-verified
//
#include <hip/hip_runtime.h>

// ---------------- problem constants (from reference) ----------------
#define B_    32
#define H_    200
#define W_    400
#define CH_   65          // CELL*CELL + 1
#define hC_   25          // H/8
#define wC_   50          // W/8
#define S_    8

// output segment offsets, in floats (flat concat in return order)
#define O_SEM 0LL
#define O_DIS 10240000LL     // + 32*4*200*400
#define O_VTX 17920000LL     // + 32*3*200*400
#define O_EMB 20520000LL     // + 32*65*25*50
#define O_DIR 61480000LL     // + 32*16*200*400
#define O_MS  156200000LL    // + 32*37*200*400
#define O_VM  158760000LL    // + 32*200*400
#define O_CO  161320000LL    // + 32*200*400
#define O_DT  161480000LL    // + 200*400*2   (total = 164,040,000)

// ---------------- CDNA5 async global<->LDS path (guarded) ----------------
#if defined(__gfx1250__)
#if __has_builtin(__builtin_amdgcn_global_load_async_to_lds_b128) && \
    __has_builtin(__builtin_amdgcn_global_store_async_from_lds_b128) && \
    __has_builtin(__builtin_amdgcn_s_wait_asynccnt)
#define VMN_USE_ASYNC 1
#endif
#endif

// Builtins take pointers to a 128-bit int vector:
//   arg0: v4i* in global (AS1), arg1: v4i* in LDS (AS3), then imm offset, imm cpol
typedef int v4i_t __attribute__((ext_vector_type(4)));
typedef __attribute__((address_space(1))) v4i_t g_v4i;   // global
typedef __attribute__((address_space(3))) v4i_t l_v4i;   // LDS

#define CHUNK 4           // float4s per thread -> 4 outstanding async loads
#define TILE4 (256 * CHUNK)

// Bulk passthrough: 4 float4s (64B) per thread, staged through LDS with the
// CDNA5 async datapath (ASYNCcnt). 625 MB of the 656 MB output goes here.
// Block-uniform fast path (no bounds checks) for full tiles; guarded tail
// path only for the at-most-one remainder block per segment.
__global__ void __launch_bounds__(256) vmn_copy(const float4* __restrict__ src,
                                                float4* __restrict__ dst,
                                                int n4, int nfullblk) {
  const int tid  = threadIdx.x;
  const int base = blockIdx.x * TILE4 + tid;
#if defined(VMN_USE_ASYNC)
  __shared__ float4 tile[TILE4];         // 16 KB of the 320 KB/WGP pool
  if ((int)blockIdx.x < nfullblk) {
    // Full tile: LDS and global chunk offsets are both k*4096 B, so each
    // chunk is one async op with a different INST_OFFSET immediate.
#pragma unroll
    for (int k = 0; k < CHUNK; ++k) {
      __builtin_amdgcn_global_load_async_to_lds_b128(
          (g_v4i*)(src + base + k * 256), (l_v4i*)&tile[tid + k * 256], 0, 0);
    }
    __builtin_amdgcn_s_wait_asynccnt(0);
#pragma unroll
    for (int k = 0; k < CHUNK; ++k) {
      __builtin_amdgcn_global_store_async_from_lds_b128(
          (g_v4i*)(dst + base + k * 256), (l_v4i*)&tile[tid + k * 256], 0, 0);
    }
    __builtin_amdgcn_s_wait_asynccnt(0);
  } else {
    // Tail tile: per-chunk bounds checks.
#pragma unroll
    for (int k = 0; k < CHUNK; ++k) {
      if (base + k * 256 < n4) {
        __builtin_amdgcn_global_load_async_to_lds_b128(
            (g_v4i*)(src + base + k * 256), (l_v4i*)&tile[tid + k * 256], 0, 0);
      }
    }
    __builtin_amdgcn_s_wait_asynccnt(0);
#pragma unroll
    for (int k = 0; k < CHUNK; ++k) {
      if (base + k * 256 < n4) {
        __builtin_amdgcn_global_store_async_from_lds_b128(
            (g_v4i*)(dst + base + k * 256), (l_v4i*)&tile[tid + k * 256], 0, 0);
      }
    }
    __builtin_amdgcn_s_wait_asynccnt(0);
  }
#else
  (void)nfullblk;
#pragma unroll
  for (int k = 0; k < CHUNK; ++k) {
    const int i = base + k * 256;
    if (i < n4) dst[i] = src[i];
  }
#endif
}

// One thread per 8x8 cell: softmax-argmax over 65 channels, then the 64-pixel
// epilogue.  Forward value of the straight-through one-hot is exactly the hard
// argmax one-hot, so masked_scores has at most one nonzero per cell = 1/sum.
__global__ void __launch_bounds__(256) vmn_cells(const float* __restrict__ vertex,
                                                 const float* __restrict__ distance,
                                                 float* __restrict__ out) {
  const int cell = blockIdx.x * 256 + threadIdx.x;
  if (cell >= B_ * hC_ * wC_) return;
  const int cx = cell % wC_;
  const int t  = cell / wC_;
  const int cy = t % hC_;
  const int b  = t / hC_;

  // vertex[b, c, cy, cx]; channel stride hC_*wC_ = 1250 (coalesced across cx)
  const float* vtx = vertex + (long long)b * (CH_ * hC_ * wC_) + cy * wC_ + cx;
  float vv[CH_];
#pragma unroll
  for (int c = 0; c < CH_; ++c) vv[c] = vtx[c * (hC_ * wC_)];

  float vmax = vv[0];
  int   arg  = 0;
#pragma unroll
  for (int c = 1; c < CH_; ++c) {
    if (vv[c] > vmax) { vmax = vv[c]; arg = c; }   // first-occurrence argmax
  }
  float sum = 0.0f;
#pragma unroll
  for (int c = 0; c < CH_; ++c) sum += __expf(vv[c] - vmax);
  const float sc = 1.0f / sum;                      // softmax score at argmax
  const float cm = (arg < CH_ - 1) ? 1.0f : 0.0f;   // cell_mask (argmax != 64)

  const int ybase = cy * S_;
  const int xbase = cx * S_;
  float* ms = out + O_MS + (long long)b * (H_ * W_);
  float* vm = out + O_VM + (long long)b * (H_ * W_);
  float* dt = out + O_DT + ((long long)(b * hC_ + cy) * wC_ + cx) * 64;
  const float* dis = distance + (long long)b * (3 * H_ * W_);

  for (int i = 0; i < 64; ++i) {
    const int iy = i >> 3, ix = i & 7;
    const int y = ybase + iy, x = xbase + ix;
    const int pix = y * W_ + x;
    const bool border = (y >= S_) && (y < H_ - S_) && (x >= S_) && (x < W_ - S_);
    const bool hit = border && (arg == i);          // arg==i implies arg<64
    ms[pix] = hit ? sc : 0.0f;
    vm[pix] = hit ? 1.0f : 0.0f;
    const float d0 = dis[pix];
    const float d1 = dis[H_ * W_ + pix];
    const float d2 = dis[2 * H_ * W_ + pix];
    const float dmx = fmaxf(d0, fmaxf(d1, d2));
    dt[i] = cm * (dmx * 0.1f);                      // / DIST_THRESHOLD
  }
}

// coords[y, x, 0] = x*0.15 - 30 ; coords[y, x, 1] = y*0.15 - 15 (batch-free)
__global__ void __launch_bounds__(256) vmn_coords(float* __restrict__ out) {
  const int idx = blockIdx.x * 256 + threadIdx.x;
  if (idx >= H_ * W_) return;
  const int x = idx % W_;
  const int y = idx / W_;
  float2 c;
  c.x = (float)x * 0.15f - 30.0f;
  c.y = (float)y * 0.15f - 15.0f;
  *(float2*)(out + O_CO + 2LL * idx) = c;           // 8B-aligned
}

extern "C" void kernel_launch(void* const* d_in, const int* in_sizes, int n_in,
                              void* d_out, int out_size, void* d_ws, size_t ws_size,
                              hipStream_t stream) {
  (void)n_in; (void)out_size; (void)d_ws; (void)ws_size;
  float* out = (float*)d_out;

  // passthroughs: semantic, distance, vertex, embedding, direction
  const long long off[5] = {O_SEM, O_DIS, O_VTX, O_EMB, O_DIR};
  for (int k = 0; k < 5; ++k) {
    const int n4 = in_sizes[k] / 4;                 // all sizes divisible by 4
    const int nfullblk = n4 / TILE4;
    const int blocks = (n4 + TILE4 - 1) / TILE4;
    vmn_copy<<<blocks, 256, 0, stream>>>((const float4*)d_in[k],
                                         (float4*)(out + off[k]), n4, nfullblk);
  }

  const int ncell = B_ * hC_ * wC_;                 // 40,000
  vmn_cells<<<(ncell + 255) / 256, 256, 0, stream>>>(
      (const float*)d_in[2] /*vertex*/, (const float*)d_in[1] /*distance*/, out);

  vmn_coords<<<(H_ * W_ + 255) / 256, 256, 0, stream>>>(out);
}